// HeteroGATLayer_63599875719680
// MI455X (gfx1250) — compile-verified
//
#include <hip/hip_runtime.h>
#include <math.h>
#include <stdint.h>

typedef float v2f __attribute__((ext_vector_type(2)));
typedef float v8f __attribute__((ext_vector_type(8)));

#define DIM 256
#define HC  64
#define NH  4
#define EDIM 768

#if defined(__AMDGCN__) && __has_builtin(__builtin_amdgcn_global_load_async_to_lds_b128)
#define HAS_ASYNC_LDS 1
#else
#define HAS_ASYNC_LDS 0
#endif

#if HAS_ASYNC_LDS
typedef int v4i_ __attribute__((vector_size(16)));
typedef __attribute__((address_space(1))) v4i_* async_gptr;
typedef __attribute__((address_space(3))) v4i_* async_lptr;
#endif

// 16-byte global -> LDS copy; async DMA path on gfx1250, plain copy fallback.
__device__ __forceinline__ void cp16(float* l, const float* g) {
#if HAS_ASYNC_LDS
  __builtin_amdgcn_global_load_async_to_lds_b128(
      (async_gptr)(uintptr_t)g,
      (async_lptr)(unsigned)(uintptr_t)l,
      0, 0);
#else
  *(float4*)l = *(const float4*)g;
#endif
}
__device__ __forceinline__ void cp_wait() {
#if HAS_ASYNC_LDS
  __builtin_amdgcn_s_wait_asynccnt(0);
#endif
}

// ---- order-preserving float <-> uint key (for atomic segment max) ----
__device__ __forceinline__ unsigned fkey(float f) {
  unsigned u = __float_as_uint(f);
  return (u & 0x80000000u) ? ~u : (u | 0x80000000u);
}
__device__ __forceinline__ float funkey(unsigned k) {
  unsigned b = (k & 0x80000000u) ? (k & 0x7FFFFFFFu) : ~k;
  return __uint_as_float(b);
}
#define NEG_INF_KEY 0x007FFFFFu  // fkey(-inf)

// ---------------- fills ----------------
__global__ void k_fill_f(float* __restrict__ p, float v, size_t n) {
  size_t gid = (size_t)blockIdx.x * blockDim.x + threadIdx.x;
  if (gid < n) p[gid] = v;
}
__global__ void k_fill_u(unsigned* __restrict__ p, unsigned v, size_t n) {
  size_t gid = (size_t)blockIdx.x * blockDim.x + threadIdx.x;
  if (gid < n) p[gid] = v;
}

// ---------------- fold a_edge into We:  w_edge[d,h] = sum_c We[d,h*64+c]*a_edge[h,c]
__global__ void k_wedge(const float* __restrict__ We0, const float* __restrict__ ae0,
                        const float* __restrict__ We1, const float* __restrict__ ae1,
                        float* __restrict__ w0, float* __restrict__ w1) {
  int gid = blockIdx.x * blockDim.x + threadIdx.x;
  if (gid >= 2 * EDIM * NH) return;
  int rel = (gid >= EDIM * NH) ? 1 : 0;
  int r = gid - rel * EDIM * NH;
  int d = r >> 2, h = r & 3;
  const float* We = rel ? We1 : We0;
  const float* ae = rel ? ae1 : ae0;
  float s = 0.f;
  #pragma unroll 4
  for (int c = 0; c < HC; ++c)
    s += We[(size_t)d * DIM + h * HC + c] * ae[h * HC + c];
  (rel ? w1 : w0)[r] = s;
}

// ---------------- node GEMM: H = X @ W  (fp32 WMMA 16x16x4) ----------------
// block = 256 threads (8 waves). Tile: 32 rows x 64 cols; each wave owns a 16x16 tile.
// Double-buffered LDS staging; async global->LDS DMA overlapped with WMMA compute.
__global__ __launch_bounds__(256) void k_gemm256(const float* __restrict__ X,
                                                 const float* __restrict__ W,
                                                 float* __restrict__ Hout, int Nrows) {
  __shared__ float As[2][32][68];
  __shared__ float Bs[2][64][68];
  const int tid   = threadIdx.x;
  const int lane  = tid & 31;
  const int wave  = tid >> 5;
  const int msub  = (wave & 1) * 16;   // 0 or 16
  const int nsub  = (wave >> 1) * 16;  // 0,16,32,48
  const int rowBase = blockIdx.x * 32;
  const int colBase = blockIdx.y * 64;
  const int l16   = lane & 15;
  const int kHalf = (lane >> 4) * 2;   // lanes 0-15 -> K+0/K+1, lanes 16-31 -> K+2/K+3

  const int lr = tid >> 4;        // 0..15
  const int lc = (tid & 15) * 4;  // 0..60

  // row indices for A staging (clamped; keeps EXEC full for WMMA)
  int arow0 = rowBase + lr;      if (arow0 >= Nrows) arow0 = Nrows - 1;
  int arow1 = rowBase + lr + 16; if (arow1 >= Nrows) arow1 = Nrows - 1;

  v8f c = {};

  // ---- stage chunk 0 into buffer 0 ----
  {
    const int k0 = 0;
    cp16(&As[0][lr][lc],      X + (size_t)arow0 * DIM + k0 + lc);
    cp16(&As[0][lr + 16][lc], X + (size_t)arow1 * DIM + k0 + lc);
    #pragma unroll
    for (int rr = lr; rr < 64; rr += 16)
      cp16(&Bs[0][rr][lc], W + (size_t)(k0 + rr) * DIM + colBase + lc);
  }
  cp_wait();
  __syncthreads();

  int buf = 0;
  for (int k0 = 0; k0 < DIM; k0 += 64) {
    // ---- issue async staging of next chunk into the other buffer ----
    if (k0 + 64 < DIM) {
      const int kn = k0 + 64;
      cp16(&As[buf ^ 1][lr][lc],      X + (size_t)arow0 * DIM + kn + lc);
      cp16(&As[buf ^ 1][lr + 16][lc], X + (size_t)arow1 * DIM + kn + lc);
      #pragma unroll
      for (int rr = lr; rr < 64; rr += 16)
        cp16(&Bs[buf ^ 1][rr][lc], W + (size_t)(kn + rr) * DIM + colBase + lc);
    }
    // ---- 16 WMMAs on current buffer ----
    #pragma unroll
    for (int kk = 0; kk < 64; kk += 4) {
      v2f a, b;
      // A 16x4 layout: lane<16 M=lane holds K,K+1 ; lane>=16 M=lane-16 holds K+2,K+3
      a.x = As[buf][msub + l16][kk + kHalf];
      a.y = As[buf][msub + l16][kk + kHalf + 1];
      // B 4x16 layout: VGPR0 rows K / K+2, VGPR1 rows K+1 / K+3, N = lane&15
      b.x = Bs[buf][kk + kHalf][nsub + l16];
      b.y = Bs[buf][kk + kHalf + 1][nsub + l16];
      c = __builtin_amdgcn_wmma_f32_16x16x4_f32(false, a, false, b, (short)0, c,
                                                false, false);
    }
    cp_wait();        // next-chunk DMA complete (this wave's ASYNCcnt)
    __syncthreads();  // all waves done reading current buffer / writing next
    buf ^= 1;
  }

  // C layout: VGPR i -> M=i (lanes 0-15), M=i+8 (lanes 16-31); N = lane&15
  const int col  = colBase + nsub + l16;
  const int row0 = rowBase + msub + (lane >> 4) * 8;
  #pragma unroll
  for (int i = 0; i < 8; ++i) {
    int row = row0 + i;
    if (row < Nrows) Hout[(size_t)row * DIM + col] = c[i];
  }
}

// ---------------- node attention scores: s_src/s_dst for both relations ----------------
__global__ void k_scores(const float* __restrict__ H0, const float* __restrict__ H1,
                         const float* __restrict__ as0, const float* __restrict__ ad0,
                         const float* __restrict__ as1, const float* __restrict__ ad1,
                         float* __restrict__ ss0, float* __restrict__ sd0,
                         float* __restrict__ ss1, float* __restrict__ sd1, int Nrows) {
  int gid = blockIdx.x * blockDim.x + threadIdx.x;
  if (gid >= Nrows * NH) return;
  int n = gid >> 2, h = gid & 3;
  const float* p0 = H0 + (size_t)n * DIM + h * HC;
  const float* p1 = H1 + (size_t)n * DIM + h * HC;
  float r00 = 0.f, r01 = 0.f, r10 = 0.f, r11 = 0.f;
  #pragma unroll 4
  for (int c = 0; c < HC; ++c) {
    float v0 = p0[c], v1 = p1[c];
    r00 += v0 * as0[h * HC + c];
    r01 += v0 * ad0[h * HC + c];
    r10 += v1 * as1[h * HC + c];
    r11 += v1 * ad1[h * HC + c];
  }
  ss0[gid] = r00; sd0[gid] = r01; ss1[gid] = r10; sd1[gid] = r11;
}

// ---------------- per-edge score: s_edge via wave dot, leaky-relu, atomic segment max ----
__global__ __launch_bounds__(256) void k_edge_alpha(const float* __restrict__ EA,
                                                    const int* __restrict__ eidx,
                                                    const float* __restrict__ wedge,
                                                    const float* __restrict__ ss,
                                                    const float* __restrict__ sd,
                                                    float* __restrict__ alpha,
                                                    unsigned* __restrict__ mkey, int E) {
  int gw   = (blockIdx.x * 256 + threadIdx.x) >> 5;  // one wave per edge
  int lane = threadIdx.x & 31;
  if (gw >= E) return;
  const float* ea = EA + (size_t)gw * EDIM;
  float a0 = 0.f, a1 = 0.f, a2 = 0.f, a3 = 0.f;
  for (int d = lane; d < EDIM; d += 32) {
    float v = ea[d];
    float4 w = *(const float4*)(wedge + d * 4);
    a0 += v * w.x; a1 += v * w.y; a2 += v * w.z; a3 += v * w.w;
  }
  #pragma unroll
  for (int s = 16; s; s >>= 1) {
    a0 += __shfl_xor(a0, s, 32);
    a1 += __shfl_xor(a1, s, 32);
    a2 += __shfl_xor(a2, s, 32);
    a3 += __shfl_xor(a3, s, 32);
  }
  if (lane == 0) {
    int src = eidx[gw], dst = eidx[E + gw];
    float se[4] = {a0, a1, a2, a3};
    #pragma unroll
    for (int h = 0; h < NH; ++h) {
      float a = ss[src * 4 + h] + sd[dst * 4 + h] + se[h];
      a = a > 0.f ? a : 0.2f * a;                    // leaky_relu(0.2)
      alpha[(size_t)gw * 4 + h] = a;
      atomicMax(&mkey[dst * 4 + h], fkey(a));        // global_atomic_max_u32
    }
  }
}

// ---------------- exp(alpha - m[dst]) + atomic denominator ----------------
__global__ void k_exp(const int* __restrict__ eidx, const unsigned* __restrict__ mkey,
                      float* __restrict__ alpha, float* __restrict__ den, int E) {
  int gid = blockIdx.x * blockDim.x + threadIdx.x;
  if (gid >= E * NH) return;
  int e = gid >> 2, h = gid & 3;
  int dst = eidx[E + e];
  float ex = __expf(alpha[gid] - funkey(mkey[dst * 4 + h]));
  alpha[gid] = ex;
  atomicAdd(&den[dst * 4 + h], ex);
}

// ---------------- message scatter: acc[dst] += h[src] * p (one block per edge) ----------
__global__ __launch_bounds__(256) void k_scatter(const float* __restrict__ Hs,
                                                 const int* __restrict__ eidx,
                                                 const float* __restrict__ exw,
                                                 const float* __restrict__ den,
                                                 float* __restrict__ acc, int E) {
  int e = blockIdx.x;
  int r = threadIdx.x;       // 0..255  (h = r/64, c = r%64)
  int h = r >> 6;
  int src = eidx[e], dst = eidx[E + e];
  float p = exw[(size_t)e * 4 + h] / den[dst * 4 + h];
  atomicAdd(&acc[(size_t)dst * DIM + r], Hs[(size_t)src * DIM + r] * p);
}

// ---------------- epilogue: relu(acc + b0 + b1) + x ----------------
__global__ void k_final(const float* __restrict__ acc, const float* __restrict__ bias0,
                        const float* __restrict__ bias1, const float* __restrict__ X,
                        float* __restrict__ out, size_t n) {
  size_t gid = (size_t)blockIdx.x * blockDim.x + threadIdx.x;
  if (gid >= n) return;
  int d = (int)(gid & (DIM - 1));
  float v = acc[gid] + bias0[d] + bias1[d];
  out[gid] = fmaxf(v, 0.f) + X[gid];
}

extern "C" void kernel_launch(void* const* d_in, const int* in_sizes, int n_in,
                              void* d_out, int out_size, void* d_ws, size_t ws_size,
                              hipStream_t stream) {
  (void)n_in; (void)out_size; (void)ws_size;
  const float* X   = (const float*)d_in[0];
  const int*   ei0 = (const int*)d_in[1];
  const int*   ei1 = (const int*)d_in[2];
  const float* EA0 = (const float*)d_in[3];
  const float* EA1 = (const float*)d_in[4];
  const float* W0  = (const float*)d_in[5];
  const float* as0 = (const float*)d_in[6];
  const float* ad0 = (const float*)d_in[7];
  const float* We0 = (const float*)d_in[8];
  const float* ae0 = (const float*)d_in[9];
  const float* b0  = (const float*)d_in[10];
  const float* W1  = (const float*)d_in[11];
  const float* as1 = (const float*)d_in[12];
  const float* ad1 = (const float*)d_in[13];
  const float* We1 = (const float*)d_in[14];
  const float* ae1 = (const float*)d_in[15];
  const float* b1  = (const float*)d_in[16];

  const int N = in_sizes[0] / DIM;
  const int E = in_sizes[1] / 2;

  // ---- workspace layout ----
  float* ws = (float*)d_ws;
  size_t fN = (size_t)N * DIM;
  float*    h0   = ws;
  float*    h1   = h0 + fN;
  float*    acc  = h1 + fN;
  float*    ss0  = acc + fN;
  float*    sd0  = ss0 + (size_t)N * NH;
  float*    ss1  = sd0 + (size_t)N * NH;
  float*    sd1  = ss1 + (size_t)N * NH;
  float*    den0 = sd1 + (size_t)N * NH;
  float*    den1 = den0 + (size_t)N * NH;
  unsigned* m0   = (unsigned*)(den1 + (size_t)N * NH);
  unsigned* m1   = m0 + (size_t)N * NH;
  float*    al0  = (float*)(m1 + (size_t)N * NH);
  float*    al1  = al0 + (size_t)E * NH;
  float*    wg0  = al1 + (size_t)E * NH;
  float*    wg1  = wg0 + (size_t)EDIM * NH;

  // ---- init accumulators (den0/den1 and m0/m1 are each contiguous) ----
  k_fill_f<<<(unsigned)((fN + 255) / 256), 256, 0, stream>>>(acc, 0.f, fN);
  k_fill_f<<<(unsigned)(((size_t)N * NH * 2 + 255) / 256), 256, 0, stream>>>(
      den0, 0.f, (size_t)N * NH * 2);
  k_fill_u<<<(unsigned)(((size_t)N * NH * 2 + 255) / 256), 256, 0, stream>>>(
      m0, NEG_INF_KEY, (size_t)N * NH * 2);

  // ---- fold a_edge into We (both relations) ----
  k_wedge<<<(2 * EDIM * NH + 255) / 256, 256, 0, stream>>>(We0, ae0, We1, ae1, wg0, wg1);

  // ---- node GEMMs via fp32 WMMA ----
  dim3 g((N + 31) / 32, DIM / 64);
  k_gemm256<<<g, 256, 0, stream>>>(X, W0, h0, N);
  k_gemm256<<<g, 256, 0, stream>>>(X, W1, h1, N);

  // ---- node attention scores ----
  k_scores<<<(N * NH + 255) / 256, 256, 0, stream>>>(h0, h1, as0, ad0, as1, ad1,
                                                     ss0, sd0, ss1, sd1, N);

  // ---- per-edge alpha + segment max ----
  int eb = (E * 32 + 255) / 256;
  k_edge_alpha<<<eb, 256, 0, stream>>>(EA0, ei0, wg0, ss0, sd0, al0, m0, E);
  k_edge_alpha<<<eb, 256, 0, stream>>>(EA1, ei1, wg1, ss1, sd1, al1, m1, E);

  // ---- exp + denominator ----
  k_exp<<<(E * NH + 255) / 256, 256, 0, stream>>>(ei0, m0, al0, den0, E);
  k_exp<<<(E * NH + 255) / 256, 256, 0, stream>>>(ei1, m1, al1, den1, E);

  // ---- weighted message scatter into shared accumulator ----
  k_scatter<<<E, 256, 0, stream>>>(h0, ei0, al0, den0, acc, E);
  k_scatter<<<E, 256, 0, stream>>>(h1, ei1, al1, den1, acc, E);

  // ---- bias + relu + residual ----
  k_final<<<(unsigned)((fN + 255) / 256), 256, 0, stream>>>(acc, b0, b1, X,
                                                            (float*)d_out, fN);
}